// HeteroRGCN_79044578115861
// MI455X (gfx1250) — compile-verified
//
#include <hip/hip_runtime.h>

// CDNA5 / gfx1250: wave32, WMMA f32 16x16x4.
typedef __attribute__((ext_vector_type(2))) float v2f;
typedef __attribute__((ext_vector_type(8))) float v8f;

__device__ __forceinline__ float lrelu01(float x) { return x > 0.0f ? x : 0.01f * x; }

// Native no-return f32 atomic add at device scope (avoids clang's CAS-loop
// expansion of atomicrmw fadd when fine-grained-memory can't be ruled out).
// No-return atomics are tracked by STOREcnt; S_ENDPGM does an implicit
// wait-idle, so results are visible to subsequent kernels on the stream.
__device__ __forceinline__ void atomAddF(float* p, float v) {
#if defined(__HIP_DEVICE_COMPILE__) && defined(__gfx1250__)
  asm volatile("global_atomic_add_f32 %0, %1, off scope:SCOPE_DEV"
               :
               : "v"(p), "v"(v)
               : "memory");
#else
  (void)__hip_atomic_fetch_add(p, v, __ATOMIC_RELAXED, __HIP_MEMORY_SCOPE_AGENT);
#endif
}

// One wave computes a 16-row output tile: D[16,16] = A[16,K] x W_r[K,16] + b_r
// for NREL relations sharing the same A. Relation 0 is stored densely to out0;
// relations 1 and 2 (SCATTER) are atomically scattered via ci/mi.
// A-fragment layout (32-bit A 16x4): lanes 0-15 hold M=lane, K={k,k+1};
// lanes 16-31 hold M=lane-16, K={k+2,k+3}. B mirrors with N=lane&15.
// C/D layout: VGPR v -> M = v + (lane>=16 ? 8 : 0), N = lane&15.
template <int K, int NREL, bool LRELU, bool SCATTER>
__launch_bounds__(256)
__global__ void rgcn_gemm(const float* __restrict__ x, int N,
                          const float* __restrict__ W, const float* __restrict__ b,
                          float* __restrict__ out0,
                          float* __restrict__ acc_c, const int* __restrict__ ci,
                          float* __restrict__ acc_m, const int* __restrict__ mi) {
  __shared__ float sW[NREL * K * 16];
  __shared__ float sB[NREL * 16];
  for (int i = threadIdx.x; i < NREL * K * 16; i += blockDim.x) sW[i] = W[i];
  for (int i = threadIdx.x; i < NREL * 16; i += blockDim.x) sB[i] = b[i];
  __syncthreads();

  const int wave = threadIdx.x >> 5;
  const int lane = threadIdx.x & 31;
  const int tile = blockIdx.x * (blockDim.x >> 5) + wave;
  if (tile * 16 >= N) return;  // wave-uniform: EXEC stays all-1s for WMMA

  const int nidx = lane & 15;         // A-row M / B,C column N
  const int koff = (lane >> 4) << 1;  // 0 (lanes 0-15) or 2 (lanes 16-31)
  const float* xrow = x + (size_t)(tile * 16 + nidx) * K;

  v8f acc[NREL];
#pragma unroll
  for (int r = 0; r < NREL; ++r) {
    const float bv = sB[r * 16 + nidx];  // bias broadcast along rows
#pragma unroll
    for (int v = 0; v < 8; ++v) acc[r][v] = bv;
  }

  for (int k = 0; k < K; k += 4) {
    float a0 = xrow[k + koff];
    float a1 = xrow[k + koff + 1];
    if (LRELU) { a0 = lrelu01(a0); a1 = lrelu01(a1); }
    v2f A;
    A.x = a0;
    A.y = a1;
#pragma unroll
    for (int r = 0; r < NREL; ++r) {
      v2f B;
      B.x = sW[r * K * 16 + (k + koff) * 16 + nidx];
      B.y = sW[r * K * 16 + (k + koff + 1) * 16 + nidx];
      acc[r] = __builtin_amdgcn_wmma_f32_16x16x4_f32(
          /*neg_a=*/false, A, /*neg_b=*/false, B,
          /*c_mod=*/(short)0, acc[r], /*reuse_a=*/false, /*reuse_b=*/false);
    }
  }

  const int mbase = tile * 16 + ((lane >> 4) << 3);  // +8 for lanes 16-31
#pragma unroll
  for (int v = 0; v < 8; ++v) out0[(size_t)(mbase + v) * 16 + nidx] = acc[0][v];

  if (SCATTER) {
#pragma unroll
    for (int v = 0; v < 8; ++v) {
      const int m = mbase + v;
      const int c = ci[m];
      atomAddF(&acc_c[(size_t)c * 16 + nidx], acc[1][v]);
      const int g = mi[m];
      atomAddF(&acc_m[(size_t)g * 16 + nidx], acc[2][v]);
    }
  }
}

__global__ void zero_f4(float4* __restrict__ p, int n4) {
  const int i = blockIdx.x * blockDim.x + threadIdx.x;
  if (i < n4) p[i] = make_float4(0.f, 0.f, 0.f, 0.f);
}

__global__ void count_edges(const int* __restrict__ ci, const int* __restrict__ mi,
                            float* __restrict__ cnt_c, float* __restrict__ cnt_m, int Nt) {
  const int i = blockIdx.x * blockDim.x + threadIdx.x;
  if (i >= Nt) return;
  atomAddF(&cnt_c[ci[i]], 1.0f);
  atomAddF(&cnt_m[mi[i]], 1.0f);
}

// h_t[i,:] += Wh_c[ci[i],:] + Wh_m[mi[i],:]   (one thread per txn-quad, b128 ops)
__global__ void gather_add(float* __restrict__ ht, const float* __restrict__ whc,
                           const float* __restrict__ whm, const int* __restrict__ ci,
                           const int* __restrict__ mi, int Nt) {
  const int idx = blockIdx.x * blockDim.x + threadIdx.x;
  if (idx >= Nt * 4) return;
  const int i = idx >> 2, q = (idx & 3) << 2;
  const int c = ci[i], m = mi[i];
  float4 s = *(const float4*)(ht + (size_t)i * 16 + q);
  const float4 a = *(const float4*)(whc + (size_t)c * 16 + q);
  const float4 bb = *(const float4*)(whm + (size_t)m * 16 + q);
  s.x += a.x + bb.x;
  s.y += a.y + bb.y;
  s.z += a.z + bb.z;
  s.w += a.w + bb.w;
  *(float4*)(ht + (size_t)i * 16 + q) = s;
}

// acc[i,:] /= max(cnt[i], 1)   (in-place -> becomes h_c / h_m)
__global__ void seg_div(float* __restrict__ acc, const float* __restrict__ cnt, int Nn) {
  const int idx = blockIdx.x * blockDim.x + threadIdx.x;
  if (idx >= Nn * 4) return;
  const int i = idx >> 2, q = (idx & 3) << 2;
  const float d = 1.0f / fmaxf(cnt[i], 1.0f);
  float4 v = *(const float4*)(acc + (size_t)i * 16 + q);
  v.x *= d;
  v.y *= d;
  v.z *= d;
  v.w *= d;
  *(float4*)(acc + (size_t)i * 16 + q) = v;
}

__global__ void out_proj(const float* __restrict__ ht, const float* __restrict__ Wout,
                         const float* __restrict__ bout, float* __restrict__ out, int Nt) {
  const int i = blockIdx.x * blockDim.x + threadIdx.x;
  if (i >= Nt) return;
  float a0 = bout[0], a1 = bout[1];
  const float* h = ht + (size_t)i * 16;
#pragma unroll
  for (int j = 0; j < 16; ++j) {
    const float v = h[j];
    a0 += v * Wout[j * 2 + 0];
    a1 += v * Wout[j * 2 + 1];
  }
  out[(size_t)i * 2 + 0] = a0;
  out[(size_t)i * 2 + 1] = a1;
}

static inline int cdiv(int a, int b) { return (a + b - 1) / b; }

extern "C" void kernel_launch(void* const* d_in, const int* in_sizes, int n_in,
                              void* d_out, int out_size, void* d_ws, size_t ws_size,
                              hipStream_t stream) {
  const float* feats = (const float*)d_in[0];
  const float* emb_c = (const float*)d_in[1];
  const float* emb_m = (const float*)d_in[2];
  const float* W0 = (const float*)d_in[3];
  const float* b0 = (const float*)d_in[4];
  const float* W1 = (const float*)d_in[5];
  const float* b1 = (const float*)d_in[6];
  const float* W2 = (const float*)d_in[7];
  const float* b2 = (const float*)d_in[8];
  const float* Wout = (const float*)d_in[9];
  const float* bout = (const float*)d_in[10];
  const int* ci = (const int*)d_in[11];
  const int* mi = (const int*)d_in[12];
  float* out = (float*)d_out;

  const int Nt = in_sizes[11];       // 1,000,000
  const int Nc = in_sizes[1] / 128;  // 500,000
  const int Nm = in_sizes[2] / 128;  // 100,000

  // Workspace carve-out (~144 MB), 256B-aligned slices.
  char* ws = (char*)d_ws;
  size_t off = 0;
  auto carve = [&](size_t bytes) -> float* {
    float* p = (float*)(ws + off);
    off = (off + bytes + 255) & ~(size_t)255;
    return p;
  };
  float* ht = carve((size_t)Nt * 16 * 4);    // Wh_s then h_t (in-place across layers)
  float* whc = carve((size_t)Nc * 16 * 4);   // Wh_c
  float* accc = carve((size_t)Nc * 16 * 4);  // segment acc -> h_c
  float* whm = carve((size_t)Nm * 16 * 4);   // Wh_m
  float* accm = carve((size_t)Nm * 16 * 4);  // segment acc -> h_m
  float* cntc = carve((size_t)Nc * 4);
  float* cntm = carve((size_t)Nm * 4);
  (void)ws_size;

  const int B = 256;
  const int gT = cdiv(Nt / 16, 8), gC = cdiv(Nc / 16, 8), gM = cdiv(Nm / 16, 8);

  // ---- edge counts (constant across layers) ----
  zero_f4<<<cdiv(Nc / 4, B), B, 0, stream>>>((float4*)cntc, Nc / 4);
  zero_f4<<<cdiv(Nm / 4, B), B, 0, stream>>>((float4*)cntm, Nm / 4);
  count_edges<<<cdiv(Nt, B), B, 0, stream>>>(ci, mi, cntc, cntm, Nt);

  // ---- layer 0 (K=128, no lrelu on inputs) ----
  rgcn_gemm<128, 1, false, false><<<gC, B, 0, stream>>>(emb_c, Nc, W0, b0, whc,
                                                        nullptr, nullptr, nullptr, nullptr);
  rgcn_gemm<128, 1, false, false><<<gM, B, 0, stream>>>(emb_m, Nm, W0 + 128 * 16, b0 + 16, whm,
                                                        nullptr, nullptr, nullptr, nullptr);
  zero_f4<<<cdiv(Nc * 4, B), B, 0, stream>>>((float4*)accc, Nc * 4);
  zero_f4<<<cdiv(Nm * 4, B), B, 0, stream>>>((float4*)accm, Nm * 4);
  rgcn_gemm<128, 3, false, true><<<gT, B, 0, stream>>>(feats, Nt, W0 + 2 * 128 * 16, b0 + 32,
                                                       ht, accc, ci, accm, mi);
  gather_add<<<cdiv(Nt * 4, B), B, 0, stream>>>(ht, whc, whm, ci, mi, Nt);
  seg_div<<<cdiv(Nc * 4, B), B, 0, stream>>>(accc, cntc, Nc);
  seg_div<<<cdiv(Nm * 4, B), B, 0, stream>>>(accm, cntm, Nm);

  // ---- layers 1..2 (K=16, lrelu on inputs) ----
  const float* Ws[2] = {W1, W2};
  const float* bs[2] = {b1, b2};
  for (int l = 0; l < 2; ++l) {
    // card/merch GEMMs read h_c/h_m (in accc/accm) BEFORE re-zeroing the accumulators
    rgcn_gemm<16, 1, true, false><<<gC, B, 0, stream>>>(accc, Nc, Ws[l], bs[l], whc,
                                                        nullptr, nullptr, nullptr, nullptr);
    rgcn_gemm<16, 1, true, false><<<gM, B, 0, stream>>>(accm, Nm, Ws[l] + 256, bs[l] + 16, whm,
                                                        nullptr, nullptr, nullptr, nullptr);
    zero_f4<<<cdiv(Nc * 4, B), B, 0, stream>>>((float4*)accc, Nc * 4);
    zero_f4<<<cdiv(Nm * 4, B), B, 0, stream>>>((float4*)accm, Nm * 4);
    rgcn_gemm<16, 3, true, true><<<gT, B, 0, stream>>>(ht, Nt, Ws[l] + 512, bs[l] + 32,
                                                       ht, accc, ci, accm, mi);
    gather_add<<<cdiv(Nt * 4, B), B, 0, stream>>>(ht, whc, whm, ci, mi, Nt);
    seg_div<<<cdiv(Nc * 4, B), B, 0, stream>>>(accc, cntc, Nc);
    seg_div<<<cdiv(Nm * 4, B), B, 0, stream>>>(accm, cntm, Nm);
  }

  // ---- output projection [16,2] ----
  out_proj<<<cdiv(Nt, B), B, 0, stream>>>(ht, Wout, bout, out, Nt);
}